// Solver_91293824844468
// MI455X (gfx1250) — compile-verified
//
#include <hip/hip_runtime.h>
#include <cstdint>
#include <cstddef>

// Anisotropic eikonal solver, MI455X (gfx1250, wave32).
//
// - Reference's softmin is degenerate (mask-to-min then exp weights), so each
//   sweep is a hard min: u_new[i] = min(u[i], min over 8 adjacencies x 4
//   lambda-candidates). 8 sequential sweeps => 8 kernel launches, ping-pong
//   between d_out's u region and d_ws (only N*4 bytes scratch needed).
// - One vertex per lane (wave32-friendly). Each lane's 128 contiguous bytes
//   of adjacency are staged into LDS with CDNA5 async DMA
//   (global_load_async_to_lds_b128, ASYNCcnt), with th:TH_LOAD_NT so the
//   128 MB/sweep zero-reuse stream does not evict the hot L2-resident gather
//   set (u 2MB + vertices 4MB + tensor_field 16MB << 192MB L2).
// - tol is only needed from the last sweep: LDS atomicMax tree -> one global
//   atomicMax per block on non-negative float bits (uint order == float order).

#define MAXV        1000.0f
#define DR_ORD      10.0f
#define DR_CUT      0.01f
#define NITER       8
#define BLOCK       256

// softplus-based clamp of x to [0,1], exactly mirroring reference _softminmax
__device__ __forceinline__ float softclamp01(float x) {
    float lo;
    if (x < 0.0f) lo = log1pf(expf(DR_ORD * fminf(x, 0.0f))) / DR_ORD;
    else          lo = x + log1pf(expf(-DR_ORD * fmaxf(x, 0.0f))) / DR_ORD;
    float lm1 = lo - 1.0f;
    float hi;
    if (lo < 1.0f) hi = lo - log1pf(expf(DR_ORD * fminf(lm1, 0.0f))) / DR_ORD;
    else           hi = 1.0f - log1pf(expf(-DR_ORD * fmaxf(lm1, 0.0f))) / DR_ORD;
    return hi;
}

__global__ void eikonal_init(float* __restrict__ u0, int N, float* __restrict__ tolOut) {
    int i = blockIdx.x * blockDim.x + threadIdx.x;
    if (i < N) u0[i] = MAXV;
    if (i == 0) *tolOut = 0.0f;
}

__global__ void eikonal_sites(float* __restrict__ u0,
                              const int* __restrict__ inds,
                              const float* __restrict__ vals, int ns) {
    int i = threadIdx.x;
    if (i < ns) u0[inds[i]] = vals[i];
}

__global__ __launch_bounds__(BLOCK)
void eikonal_iter(const float* __restrict__ u_old,
                  float* __restrict__ u_new,
                  const float2* __restrict__ verts,
                  const float4* __restrict__ tf,     // tensor_field as (m00,m01,m10,m11)
                  const int4*  __restrict__ adj,     // N*8 rows of (i,j,k,s)
                  int N, int computeTol, float* __restrict__ tolOut)
{
    __shared__ int4 sAdj[BLOCK * 8];       // 32 KB: 8 rows x 16 B per lane
    __shared__ unsigned sTol;

    int gid = blockIdx.x * blockDim.x + threadIdx.x;
    int i   = (gid < N) ? gid : (N - 1);   // clamp: keep EXEC full for async DMA

    // ---- CDNA5 async DMA: stage this lane's 128 B adjacency block into LDS.
    // Non-temporal: the adjacency stream has zero reuse within a sweep and
    // must not evict the hot u/vertices/tensor_field working set from L2.
    {
        unsigned ldsa = (unsigned)(uintptr_t)(&sAdj[threadIdx.x * 8]);
        uint64_t ga   = (uint64_t)(uintptr_t)(adj + (size_t)i * 8);
#define ASYNC_LD16(OFF)                                                        \
        asm volatile("global_load_async_to_lds_b128 %0, %1, off offset:" #OFF  \
                     " th:TH_LOAD_NT"                                          \
                     :: "v"(ldsa), "v"(ga) : "memory")
        ASYNC_LD16(0);   ASYNC_LD16(16);  ASYNC_LD16(32);  ASYNC_LD16(48);
        ASYNC_LD16(64);  ASYNC_LD16(80);  ASYNC_LD16(96);  ASYNC_LD16(112);
#undef ASYNC_LD16
        asm volatile("s_wait_asynccnt 0" ::: "memory");
    }

    float  ui   = u_old[i];
    float  best = ui;                       // cand[0] = self value
    float2 vi   = verts[i];
    const int4* rows = &sAdj[threadIdx.x * 8];

#pragma unroll
    for (int a = 0; a < 8; ++a) {
        int4 r = rows[a];
        int j = r.y, k = r.z, s = r.w;

        float  u1 = u_old[j];
        float  u2 = u_old[k];
        float2 vj = verts[j];
        float2 vk = verts[k];
        float4 M  = tf[s];

        float e1x = vi.x - vj.x, e1y = vi.y - vj.y;
        float e2x = vi.x - vk.x, e2y = vi.y - vk.y;
        float ax  = e1x - e2x,   ay  = e1y - e2y;

        float Max = M.x * ax + M.y * ay;
        float May = M.z * ax + M.w * ay;
        float Mex = M.x * e2x + M.y * e2y;
        float Mey = M.z * e2x + M.w * e2y;

        float A_ = ax * Max + ay * May;
        float B_ = ax * Mex + ay * Mey;
        float C_ = e2x * Mex + e2y * Mey;

        float d     = u1 - u2;
        float denom = A_ - d * d;
        float dsafe = (fabsf(denom) > 1e-12f) ? denom : 1.0f;
        float rad   = (A_ * C_ - B_ * B_) / dsafe;
        bool  valid = (denom > 0.0f) && (rad >= 0.0f);
        float sq    = sqrtf(valid ? rad : 0.0f);
        float A_s   = (fabsf(A_) > 1e-12f) ? A_ : 1.0f;
        float ad    = fabsf(d);
        float lam1  = (-B_ + ad * sq) / A_s;
        float lam2  = (-B_ - ad * sq) / A_s;

        // lambda = 0 candidate: t = e2, q = C_
        {
            float c0 = u2 + ((C_ > 0.0f) ? sqrtf(C_) : 0.0f);
            best = fminf(best, c0);
        }
        // lambda = 1 candidate: t = e1
        {
            float M1x = M.x * e1x + M.y * e1y;
            float M1y = M.z * e1x + M.w * e1y;
            float q1  = e1x * M1x + e1y * M1y;
            float c1  = u1 + ((q1 > 0.0f) ? sqrtf(q1) : 0.0f);
            best = fminf(best, c1);
        }
        // root candidates (soft-clamped), gated exactly like the reference
        if (valid) {
            if ((lam1 > -DR_CUT) && (lam1 < 1.0f + DR_CUT)) {
                float s1  = softclamp01(lam1);
                float tx  = s1 * e1x + (1.0f - s1) * e2x;
                float ty  = s1 * e1y + (1.0f - s1) * e2y;
                float Mtx = M.x * tx + M.y * ty;
                float Mty = M.z * tx + M.w * ty;
                float q   = tx * Mtx + ty * Mty;
                float c   = s1 * u1 + (1.0f - s1) * u2 + ((q > 0.0f) ? sqrtf(q) : 0.0f);
                best = fminf(best, c);
            }
            if ((lam2 != lam1) && (lam2 > -DR_CUT) && (lam2 < 1.0f + DR_CUT)) {
                float s2  = softclamp01(lam2);
                float tx  = s2 * e1x + (1.0f - s2) * e2x;
                float ty  = s2 * e1y + (1.0f - s2) * e2y;
                float Mtx = M.x * tx + M.y * ty;
                float Mty = M.z * tx + M.w * ty;
                float q   = tx * Mtx + ty * Mty;
                float c   = s2 * u1 + (1.0f - s2) * u2 + ((q > 0.0f) ? sqrtf(q) : 0.0f);
                best = fminf(best, c);
            }
        }
    }

    if (gid < N) u_new[gid] = best;

    if (computeTol) {
        if (threadIdx.x == 0) sTol = 0u;
        __syncthreads();
        unsigned bits = (gid < N) ? __float_as_uint(fabsf(best - ui)) : 0u;
        atomicMax(&sTol, bits);               // LDS atomic; |diff| >= 0 so uint order == float order
        __syncthreads();
        if (threadIdx.x == 0) atomicMax((unsigned*)tolOut, sTol);
    }
}

extern "C" void kernel_launch(void* const* d_in, const int* in_sizes, int n_in,
                              void* d_out, int out_size, void* d_ws, size_t ws_size,
                              hipStream_t stream) {
    const float2* verts = (const float2*)d_in[0];     // (N,2) f32
    const float4* tf    = (const float4*)d_in[1];     // (S,2,2) f32
    const int4*   adj   = (const int4*)d_in[2];       // (N,8,4) i32
    const int*    iidx  = (const int*)d_in[3];        // (n_sites,) i32
    const float*  ivals = (const float*)d_in[4];      // (n_sites,) f32

    int N  = in_sizes[0] / 2;
    int ns = in_sizes[3];

    float* uout   = (float*)d_out;          // d_out = [u (N floats), tol (1 float)]
    float* tolOut = uout + N;
    float* bufA   = uout;                   // ping-pong A = d_out's u region
    float* bufB   = (float*)d_ws;           // ping-pong B = scratch (N floats)

    int blocks = (N + BLOCK - 1) / BLOCK;

    eikonal_init<<<blocks, BLOCK, 0, stream>>>(bufA, N, tolOut);
    eikonal_sites<<<1, 64, 0, stream>>>(bufA, iidx, ivals, ns);

    float* src = bufA;
    for (int it = 0; it < NITER; ++it) {            // even count => final lands in bufA = d_out
        int   last = (it == NITER - 1) ? 1 : 0;
        float* dst = (src == bufA) ? bufB : bufA;
        eikonal_iter<<<blocks, BLOCK, 0, stream>>>(src, dst, verts, tf, adj, N, last, tolOut);
        src = dst;
    }
}